// GNN_67362267070571
// MI455X (gfx1250) — compile-verified
//
#include <hip/hip_runtime.h>
#include <hip/hip_bf16.h>

typedef float v2f __attribute__((ext_vector_type(2)));
typedef float v8f __attribute__((ext_vector_type(8)));

#define D 64

// ---------------------------------------------------------------------------
// deg[i] = 1.0 (self loop contribution)
__global__ __launch_bounds__(256) void gcn_init_deg(float* __restrict__ deg, int n) {
    int i = blockIdx.x * blockDim.x + threadIdx.x;
    if (i < n) deg[i] = 1.0f;
}

// deg[dst[e]] += 1 per edge
__global__ __launch_bounds__(256) void gcn_deg_accum(float* __restrict__ deg,
                                                     const int* __restrict__ dst, int e_cnt) {
    int e = blockIdx.x * blockDim.x + threadIdx.x;
    if (e < e_cnt) atomicAdd(&deg[dst[e]], 1.0f);
}

// dinv[i] = rsqrt(deg[i])
__global__ __launch_bounds__(256) void gcn_dinv(float* __restrict__ dinv,
                                                const float* __restrict__ deg, int n) {
    int i = blockIdx.x * blockDim.x + threadIdx.x;
    if (i < n) dinv[i] = __frsqrt_rn(deg[i]);
}

// ---------------------------------------------------------------------------
// H[N x 64] = X[N x 64] @ W[64 x 64], fp32 WMMA 16x16x4.
// One wave per 16x16 output tile. Block = 256 thr = 8 waves = 2 row-tiles x 4 col-tiles.
// N is a multiple of 16 here (100000 = 6250*16).
__global__ __launch_bounds__(256) void gcn_gemm64_wmma(const float* __restrict__ X,
                                                       const float* __restrict__ W,
                                                       float* __restrict__ H, int n_nodes) {
    const int lane = threadIdx.x & 31;
    const int wave = threadIdx.x >> 5;            // 0..7
    const int rt   = blockIdx.x * 2 + (wave >> 2);// row tile index
    const int ct   = wave & 3;                    // col tile 0..3
    const int row0 = rt * 16;
    if (row0 >= n_nodes) return;                  // uniform per wave -> EXEC stays all-1s

    const int m     = lane & 15;                  // M (A/D row group), also N for B/D
    const int khalf = (lane >> 4) * 2;            // 0 for lanes 0-15, 2 for lanes 16-31

    v8f c = {0.f, 0.f, 0.f, 0.f, 0.f, 0.f, 0.f, 0.f};

    const float* xrow = X + (size_t)(row0 + m) * D;   // A row for this lane
    const float* wcol = W + ct * 16 + m;              // B column, stride D per K row

#pragma unroll
    for (int kk = 0; kk < 16; ++kk) {
        const int kb = kk * 4 + khalf;
        // A fragment: lane holds A[m][kb], A[m][kb+1]  (16x4 tile, ISA layout)
        v2f a = *(const v2f*)(xrow + kb);
        // B fragment: lane holds B[kb][col], B[kb+1][col] (4x16 tile)
        v2f b;
        b.x = wcol[(size_t)(kb + 0) * D];
        b.y = wcol[(size_t)(kb + 1) * D];
        c = __builtin_amdgcn_wmma_f32_16x16x4_f32(false, a, false, b, (short)0, c,
                                                  false, false);
    }

    // D tile store: VGPR i -> row (i + 8*(lane>>4)), col (lane&15)
    const int mo = (lane >> 4) * 8;
    float* hout = H + (size_t)(row0 + mo) * D + ct * 16 + (lane & 15);
#pragma unroll
    for (int i = 0; i < 8; ++i) hout[(size_t)i * D] = c[i];
}

// ---------------------------------------------------------------------------
// Edge scatter: out[dst] += dinv[src]*dinv[dst] * H[src]. 32 threads per edge,
// each thread owns a float2 of the 64-wide feature.
__global__ __launch_bounds__(256) void gcn_scatter(const float* __restrict__ H,
                                                   const float* __restrict__ dinv,
                                                   const int* __restrict__ src,
                                                   const int* __restrict__ dst,
                                                   float* __restrict__ out, int e_cnt) {
    long long t = (long long)blockIdx.x * blockDim.x + threadIdx.x;
    int e = (int)(t >> 5);
    if (e >= e_cnt) return;
    int d2 = ((int)t & 31) * 2;
    int s  = src[e];
    int dd = dst[e];
    float nrm = dinv[s] * dinv[dd];
    v2f hv = *(const v2f*)(H + (size_t)s * D + d2);
    float* o = out + (size_t)dd * D + d2;
    atomicAdd(o + 0, hv.x * nrm);
    atomicAdd(o + 1, hv.y * nrm);
}

// ---------------------------------------------------------------------------
// Epilogue: agg = [relu](agg + dinv[i]^2 * H[i] + bias)   (self-loop folded in)
__global__ __launch_bounds__(256) void gcn_epilogue(float* __restrict__ agg,
                                                    const float* __restrict__ H,
                                                    const float* __restrict__ dinv,
                                                    const float* __restrict__ bias,
                                                    int n_nodes, int do_relu) {
    long long t = (long long)blockIdx.x * blockDim.x + threadIdx.x;
    long long total = (long long)n_nodes * D;
    if (t >= total) return;
    int i = (int)(t >> 6);
    int d = (int)t & 63;
    float di = dinv[i];
    float v = agg[t] + di * di * H[t] + bias[d];
    agg[t] = do_relu ? fmaxf(v, 0.0f) : v;
}

// ---------------------------------------------------------------------------
extern "C" void kernel_launch(void* const* d_in, const int* in_sizes, int n_in,
                              void* d_out, int out_size, void* d_ws, size_t ws_size,
                              hipStream_t stream) {
    const float* x   = (const float*)d_in[0];
    const int*   ei  = (const int*)d_in[1];   // [2, E]
    const float* W1  = (const float*)d_in[2];
    const float* b1  = (const float*)d_in[3];
    const float* W2  = (const float*)d_in[4];
    const float* b2  = (const float*)d_in[5];

    const int N = in_sizes[0] / D;            // 100000
    const int E = in_sizes[1] / 2;            // 1600000
    const int* src = ei;
    const int* dst = ei + E;

    float* deg  = (float*)d_ws;                      // N
    float* dinv = deg + N;                           // N
    float* h    = dinv + N;                          // N*D
    float* agg  = h + (size_t)N * D;                 // N*D
    float* out  = (float*)d_out;

    const int TB = 256;
    dim3 blk(TB);

    // degree + normalization
    gcn_init_deg<<<(N + TB - 1) / TB, blk, 0, stream>>>(deg, N);
    gcn_deg_accum<<<(E + TB - 1) / TB, blk, 0, stream>>>(deg, dst, E);
    gcn_dinv<<<(N + TB - 1) / TB, blk, 0, stream>>>(dinv, deg, N);

    const int gemm_blocks = (N + 31) / 32;           // 2 row-tiles per block
    const long long sc_threads = (long long)E * 32;
    const int sc_blocks = (int)((sc_threads + TB - 1) / TB);
    const long long ep_threads = (long long)N * D;
    const int ep_blocks = (int)((ep_threads + TB - 1) / TB);

    // ---- layer 1 ----
    gcn_gemm64_wmma<<<gemm_blocks, blk, 0, stream>>>(x, W1, h, N);
    hipMemsetAsync(agg, 0, (size_t)N * D * sizeof(float), stream);
    gcn_scatter<<<sc_blocks, blk, 0, stream>>>(h, dinv, src, dst, agg, E);
    gcn_epilogue<<<ep_blocks, blk, 0, stream>>>(agg, h, dinv, b1, N, 1);

    // ---- layer 2 ----
    gcn_gemm64_wmma<<<gemm_blocks, blk, 0, stream>>>(agg, W2, h, N);
    hipMemsetAsync(out, 0, (size_t)N * D * sizeof(float), stream);
    gcn_scatter<<<sc_blocks, blk, 0, stream>>>(h, dinv, src, dst, out, E);
    gcn_epilogue<<<ep_blocks, blk, 0, stream>>>(out, h, dinv, b2, N, 0);
}